// NeuroRouteGNN_76510547411131
// MI455X (gfx1250) — compile-verified
//
#include <hip/hip_runtime.h>
#include <hip/hip_bf16.h>

typedef float v2f __attribute__((ext_vector_type(2)));
typedef float v8f __attribute__((ext_vector_type(8)));

#define N_NODES_C 100000
#define N_EDGES_C 1600000
#define N_PAIRS_C 500000
#define HID_C 64

// ---------------- degree / normalization ----------------

__global__ void k_init_deg(float* __restrict__ deg) {
  int i = blockIdx.x * blockDim.x + threadIdx.x;
  if (i < N_NODES_C) deg[i] = 1.0f;   // +1 self-loop folded in
}

__global__ void k_deg_scatter(const int* __restrict__ dst,
                              const float* __restrict__ ew,
                              float* __restrict__ deg) {
  int e = blockIdx.x * blockDim.x + threadIdx.x;
  if (e < N_EDGES_C) atomicAdd(&deg[dst[e]], ew[e]);
}

__global__ void k_dinv(float* __restrict__ deg) {
  int i = blockIdx.x * blockDim.x + threadIdx.x;
  if (i < N_NODES_C) deg[i] = rsqrtf(deg[i]);   // in-place: deg -> dinv
}

// ---------------- dense GEMM  O[M,64] = (relu?)A[M,64] x W[64,64] ----------------
// One wave per 16-row tile, 4 f32 16x16 accumulators (all 64 cols), K loop of 16
// V_WMMA_F32_16X16X4_F32 steps.

__global__ void k_gemm64(const float* __restrict__ A, const float* __restrict__ W,
                         float* __restrict__ O, int M, int relu_in) {
  const int lane = threadIdx.x & 31;
  const int wave = threadIdx.x >> 5;
  const int tile = blockIdx.x * 8 + wave;
  const int ntiles = M >> 4;
  if (tile >= ntiles) return;            // wave-uniform: EXEC all-ones below
  const int m0   = tile << 4;
  const int l16  = lane & 15;
  const int half = lane >> 4;
  const float* arow = A + (size_t)(m0 + l16) * HID_C;

  v8f acc[4] = {};
#pragma unroll
  for (int k = 0; k < 16; ++k) {
    const int kk = 4 * k + 2 * half;                 // this lane's K pair
    v2f a = *(const v2f*)(arow + kk);
    if (relu_in) { a.x = fmaxf(a.x, 0.f); a.y = fmaxf(a.y, 0.f); }
#pragma unroll
    for (int n = 0; n < 4; ++n) {
      const int col = n * 16 + l16;
      v2f b;
      b.x = W[kk * HID_C + col];
      b.y = W[(kk + 1) * HID_C + col];
      acc[n] = __builtin_amdgcn_wmma_f32_16x16x4_f32(
          false, a, false, b, (short)0, acc[n], false, false);
    }
  }
#pragma unroll
  for (int n = 0; n < 4; ++n) {
    const int col = n * 16 + l16;
#pragma unroll
    for (int v = 0; v < 8; ++v)
      O[(size_t)(m0 + v + 8 * half) * HID_C + col] = acc[n][v];
  }
}

// ---------------- GCN aggregation ----------------

__global__ void k_agg_init(const float* __restrict__ xw, const float* __restrict__ dinv,
                           const float* __restrict__ b, float* __restrict__ out) {
  long idx = (long)blockIdx.x * blockDim.x + threadIdx.x;
  if (idx < (long)N_NODES_C * HID_C) {
    int i = (int)(idx >> 6), f = (int)(idx & 63);
    float di = dinv[i];
    out[idx] = xw[idx] * di * di + b[f];   // self-loop term + bias
  }
}

__global__ void k_edge_scatter(const int* __restrict__ src, const int* __restrict__ dst,
                               const float* __restrict__ ew, const float* __restrict__ dinv,
                               const float* __restrict__ XW, float* __restrict__ OUT) {
  int e = blockIdx.x * 4 + threadIdx.y;   // blockDim = (64,4)
  int f = threadIdx.x;                    // feature lane 0..63
  if (e >= N_EDGES_C) return;
  int s = src[e], d = dst[e];
  float norm = dinv[s] * ew[e] * dinv[d];
  atomicAdd(&OUT[(size_t)d * HID_C + f], XW[(size_t)s * HID_C + f] * norm);
}

// ---------------- fused pair-gather + MLP ----------------
// feat[p] = [relu(h[oc]), relu(h[od]), ea]  (K=129)
// hidden = relu(feat@Wf1 + bf1); q = hidden@Wf2 + bf2
// K=128 done via WMMA with gathered A rows; 129th column folded into the
// accumulator as a rank-1 VALU update; Wf2 reduction via shfl_xor.

__global__ void k_final(const float* __restrict__ H,
                        const int* __restrict__ ptr, const int* __restrict__ gidx,
                        const int* __restrict__ curr, const int* __restrict__ dstl,
                        const float* __restrict__ ea,
                        const float* __restrict__ Wf1, const float* __restrict__ bf1,
                        const float* __restrict__ Wf2, const float* __restrict__ bf2,
                        float* __restrict__ q) {
  const int lane = threadIdx.x & 31;
  const int wave = threadIdx.x >> 5;
  const int tile = blockIdx.x * 8 + wave;
  const int ntiles = N_PAIRS_C / 16;
  if (tile >= ntiles) return;            // wave-uniform
  const int p0   = tile << 4;
  const int l16  = lane & 15;
  const int half = lane >> 4;

  const int prow = p0 + l16;             // this lane's A-matrix row
  const int off  = ptr[gidx[prow]];
  const float* rc = H + (size_t)(off + curr[prow]) * HID_C;
  const float* rd = H + (size_t)(off + dstl[prow]) * HID_C;

  v8f acc[4] = {};
#pragma unroll
  for (int k = 0; k < 32; ++k) {
    const int kk = 4 * k + 2 * half;                     // 0..127
    const float* srcp = (k < 16) ? (rc + kk) : (rd + (kk - 64));
    v2f a = *(const v2f*)srcp;
    a.x = fmaxf(a.x, 0.f); a.y = fmaxf(a.y, 0.f);        // relu(h) fused here
#pragma unroll
    for (int n = 0; n < 4; ++n) {
      const int col = n * 16 + l16;
      v2f b;
      b.x = Wf1[kk * HID_C + col];
      b.y = Wf1[(kk + 1) * HID_C + col];
      acc[n] = __builtin_amdgcn_wmma_f32_16x16x4_f32(
          false, a, false, b, (short)0, acc[n], false, false);
    }
  }

  float wf2c[4], wlast[4], bb[4];
#pragma unroll
  for (int n = 0; n < 4; ++n) {
    const int col = n * 16 + l16;
    wf2c[n]  = Wf2[col];
    wlast[n] = Wf1[128 * HID_C + col];   // K=129th row of Wf1
    bb[n]    = bf1[col];
  }
  const float b2v = bf2[0];

#pragma unroll
  for (int v = 0; v < 8; ++v) {
    const int rowv = p0 + v + 8 * half;  // C-layout row this lane holds
    const float ev = ea[rowv];
    float part = 0.f;
#pragma unroll
    for (int n = 0; n < 4; ++n) {
      float hv = acc[n][v] + ev * wlast[n] + bb[n];
      part += fmaxf(hv, 0.f) * wf2c[n];
    }
#pragma unroll
    for (int m = 1; m < 16; m <<= 1)     // reduce 16 cols within each half-wave
      part += __shfl_xor(part, m, 32);
    if (l16 == 0) q[rowv] = part + b2v;
  }
}

// ---------------- launch ----------------

extern "C" void kernel_launch(void* const* d_in, const int* in_sizes, int n_in,
                              void* d_out, int out_size, void* d_ws, size_t ws_size,
                              hipStream_t stream) {
  const float* x    = (const float*)d_in[0];
  const int*   eidx = (const int*)d_in[1];
  const int*   src  = eidx;
  const int*   dst  = eidx + N_EDGES_C;
  const float* ew   = (const float*)d_in[2];
  const int*   ptr  = (const int*)d_in[3];
  const int*   gix  = (const int*)d_in[4];
  const int*   curr = (const int*)d_in[5];
  const int*   dstl = (const int*)d_in[6];
  const float* ea   = (const float*)d_in[7];
  const float* W1   = (const float*)d_in[8];
  const float* b1   = (const float*)d_in[9];
  const float* W2   = (const float*)d_in[10];
  const float* b2   = (const float*)d_in[11];
  const float* Wf1  = (const float*)d_in[12];
  const float* bf1  = (const float*)d_in[13];
  const float* Wf2  = (const float*)d_in[14];
  const float* bf2  = (const float*)d_in[15];
  float* q = (float*)d_out;

  float* ws   = (float*)d_ws;
  float* dinv = ws;                                   // N_NODES floats (deg->dinv)
  float* xw   = ws + ((N_NODES_C + 255) & ~255);      // 100000x64
  float* agg  = xw + (size_t)N_NODES_C * HID_C;       // 100000x64

  const int ntilesN = N_NODES_C / 16;                 // 6250
  const int ntilesP = N_PAIRS_C / 16;                 // 31250
  dim3 eb(64, 4, 1);

  // normalization (shared by both layers)
  k_init_deg   <<<(N_NODES_C + 255) / 256, 256, 0, stream>>>(dinv);
  k_deg_scatter<<<(N_EDGES_C + 255) / 256, 256, 0, stream>>>(dst, ew, dinv);
  k_dinv       <<<(N_NODES_C + 255) / 256, 256, 0, stream>>>(dinv);

  // GCN layer 1
  k_gemm64<<<(ntilesN + 7) / 8, 256, 0, stream>>>(x, W1, xw, N_NODES_C, 0);
  k_agg_init<<<(int)(((long)N_NODES_C * HID_C + 255) / 256), 256, 0, stream>>>(xw, dinv, b1, agg);
  k_edge_scatter<<<(N_EDGES_C + 3) / 4, eb, 0, stream>>>(src, dst, ew, dinv, xw, agg);

  // GCN layer 2 (relu of layer-1 output fused into GEMM A-load)
  k_gemm64<<<(ntilesN + 7) / 8, 256, 0, stream>>>(agg, W2, xw, N_NODES_C, 1);
  k_agg_init<<<(int)(((long)N_NODES_C * HID_C + 255) / 256), 256, 0, stream>>>(xw, dinv, b2, agg);
  k_edge_scatter<<<(N_EDGES_C + 3) / 4, eb, 0, stream>>>(src, dst, ew, dinv, xw, agg);

  // fused pair gather + MLP (relu of layer-2 output fused into gather)
  k_final<<<(ntilesP + 7) / 8, 256, 0, stream>>>(agg, ptr, gix, curr, dstl, ea,
                                                 Wf1, bf1, Wf2, bf2, q);
}